// UnitaryLinear_21096879358600
// MI455X (gfx1250) — compile-verified
//
#include <hip/hip_runtime.h>
#include <hip/hip_bf16.h>
#include <stdint.h>

// RoPE rotation, memory-bound streaming kernel for gfx1250 (MI455X).
//   x:      (8, 2048, 2048, 2) f32  -> 256 MiB in, 256 MiB out
//   angles: (1024,) f32
// 2*NR == D, so the whole tensor is rotated (no tail).
// Each contiguous 16-byte group {a0,a1,b0,b1} -> {a*c-b*s (x2), a*s+b*c (x2)}.
// Roofline: 512 MiB / 23.3 TB/s ~= 23 us; ~0.2 GFLOP total -> pure bandwidth.

typedef float fvec4 __attribute__((ext_vector_type(4)));

#define NR    1024        // number of rotations (== angle count)
#define BLOCK 256         // 8 wave32s
#define GPT   8           // float4 groups per thread
#define GPB   (BLOCK*GPT) // 2048 groups per block (multiple of NR)

__global__ __launch_bounds__(BLOCK)
void rope_rotate_kernel(const float* __restrict__ x,
                        const float* __restrict__ angles,
                        float* __restrict__ out,
                        long long ngroups)
{
    // One contiguous LDS region: [angles | cos | sin], 12 KB total.
    __shared__ __align__(16) float s_ang[NR];
    __shared__ float s_cos[NR];
    __shared__ float s_sin[NR];

    const int tid = threadIdx.x;

    // ---- CDNA5 async copy: stage the 4 KB angle table global -> LDS ----
    // 1024 floats / 256 lanes = one b128 per lane. Tracked by ASYNCcnt.
    {
        uint32_t lds_off = (uint32_t)(uintptr_t)(&s_ang[tid * 4]);
        uint64_t gaddr   = (uint64_t)(uintptr_t)(angles + tid * 4);
        asm volatile("global_load_async_to_lds_b128 %0, %1, off"
                     :: "v"(lds_off), "v"(gaddr) : "memory");
        asm volatile("s_wait_asynccnt 0x0" ::: "memory");
    }
    __syncthreads();

    // One sincos per table entry per block (hardware v_sin/v_cos),
    // fixed trip count -> no exec-mask loop machinery.
#pragma unroll
    for (int t = 0; t < NR / BLOCK; ++t) {
        const int j = t * BLOCK + tid;
        float sn, cs;
        __sincosf(s_ang[j], &sn, &cs);
        s_cos[j] = cs;
        s_sin[j] = sn;
    }
    __syncthreads();

    const fvec4* __restrict__ pin  = (const fvec4*)x;
    fvec4*       __restrict__ pout = (fvec4*)out;

    const long long base = (long long)blockIdx.x * GPB;

    if (base + GPB <= ngroups) {
        // Fast path: full block, no bounds checks -> clean b128 NT clauses.
#pragma unroll
        for (int i = 0; i < GPT; ++i) {
            const long long gi = base + (long long)(i * BLOCK + tid);
            const int   r  = (i * BLOCK + tid) & (NR - 1);
            const float cs = s_cos[r];
            const float sn = s_sin[r];

            fvec4 v = __builtin_nontemporal_load(&pin[gi]);
            fvec4 o;
            o.x = v.x * cs - v.z * sn;   // na0
            o.y = v.y * cs - v.w * sn;   // na1
            o.z = v.x * sn + v.z * cs;   // nb0
            o.w = v.y * sn + v.w * cs;   // nb1
            __builtin_nontemporal_store(o, &pout[gi]);
        }
    } else {
        // Tail block (unused for the reference shapes, kept for safety).
        for (int i = 0; i < GPT; ++i) {
            const long long gi = base + (long long)(i * BLOCK + tid);
            if (gi < ngroups) {
                const int   r  = (i * BLOCK + tid) & (NR - 1);
                const float cs = s_cos[r];
                const float sn = s_sin[r];

                fvec4 v = __builtin_nontemporal_load(&pin[gi]);
                fvec4 o;
                o.x = v.x * cs - v.z * sn;
                o.y = v.y * cs - v.w * sn;
                o.z = v.x * sn + v.z * cs;
                o.w = v.y * sn + v.w * cs;
                __builtin_nontemporal_store(o, &pout[gi]);
            }
        }
    }
}

extern "C" void kernel_launch(void* const* d_in, const int* in_sizes, int n_in,
                              void* d_out, int out_size, void* d_ws, size_t ws_size,
                              hipStream_t stream) {
    const float* x      = (const float*)d_in[0];
    const float* angles = (const float*)d_in[1];
    float*       out    = (float*)d_out;

    const long long ngroups = (long long)in_sizes[0] / 4;       // 16,777,216
    const int       blocks  = (int)((ngroups + GPB - 1) / GPB); // 8192

    rope_rotate_kernel<<<blocks, BLOCK, 0, stream>>>(x, angles, out, ngroups);
}